// KernelCRPS_84782654423782
// MI455X (gfx1250) — compile-verified
//
#include <hip/hip_runtime.h>
#include <stdint.h>

// Problem constants (from reference): B=2, E=16, L=40320, N=64
#define B_      2
#define E_      16
#define L_      40320
#define N_      64
#define LN_     (L_ * N_)          // 2,580,480 groups per batch
#define ELN_    (E_ * LN_)         // 41,287,680 elements per batch of pred
#define GROUPS_ (B_ * LN_)         // 5,160,960 total (b,l,n) groups
#define TILE_   256                // groups per tile (one per thread)
#define NTILES_ (GROUPS_ / TILE_)  // 20,160 (exact)
#define NBLK_   2520               // persistent blocks; 8 tiles per block (exact)

// -------- deterministic block tree-reduction (256 threads) ----------
__device__ __forceinline__ float block_reduce_256(float v) {
    __shared__ float red[256];
    const int tid = threadIdx.x;
    red[tid] = v;
    __syncthreads();
    #pragma unroll
    for (int s = 128; s > 0; s >>= 1) {
        if (tid < s) red[tid] += red[tid + s];
        __syncthreads();
    }
    return red[0];
}

// -------- kernel 0: W = sum(node_weights) ---------------------------
__global__ void __launch_bounds__(256)
k_nodew_sum(const float* __restrict__ nodew, float* __restrict__ wsum) {
    float s = 0.f;
    for (int i = threadIdx.x; i < L_; i += 256) s += nodew[i];
    float t = block_reduce_256(s);
    if (threadIdx.x == 0) wsum[0] = t;
}

// -------- kernel 1: main streaming CRPS reduction -------------------
// Each thread handles one (b,l,n) group per tile: 16 ensemble values,
// staged through LDS via CDNA5 async global->LDS loads (non-temporal,
// data is single-use and 1.7x the 192MB L2), double-buffered on ASYNCcnt.
// No barriers needed: each wave reads only the LDS slots it wrote.
__global__ void __launch_bounds__(256)
k_crps_main(const float* __restrict__ pred,
            const float* __restrict__ target,
            const float* __restrict__ nodew,
            const float* __restrict__ featw,
            float* __restrict__ partials) {
    __shared__ float stage[2][E_ * TILE_];   // 2 x 16 KB double buffer
    const int tid = threadIdx.x;
    float acc = 0.f;

    // n = gid % 64 is tile-invariant (tiles advance by 256, 64 | 256).
    const float fw = featw[tid & 63] * (1.0f / (float)N_);

    // Issue 16 async global->LDS loads for tile t into buffer `buf`.
    auto issue = [&](int t, int buf) {
        const unsigned gid = (unsigned)t * TILE_ + (unsigned)tid;
        const unsigned b   = gid / (unsigned)LN_;          // 0 or 1
        const unsigned r   = gid - b * (unsigned)LN_;      // l*64 + n
        const unsigned byteOff = (b * (unsigned)ELN_ + r) * 4u;  // < 2^31
        const unsigned ldsAddr = (unsigned)(uintptr_t)&stage[buf][tid];
        #pragma unroll
        for (int e = 0; e < E_; ++e) {
            // GVS mode: addr = SGPR64(pred) + VGPR32(byte offset); streaming NT
            asm volatile("global_load_async_to_lds_b32 %0, %1, %2 th:TH_LOAD_NT"
                         :: "v"(ldsAddr + (unsigned)(e * TILE_ * 4)),
                            "v"(byteOff + (unsigned)e * (unsigned)(LN_ * 4)),
                            "s"(pred)
                         : "memory");
        }
    };

    // Compute one group from buffer `buf`.
    auto compute = [&](int t, int buf) {
        const unsigned gid = (unsigned)t * TILE_ + (unsigned)tid;
        const unsigned b   = gid / (unsigned)LN_;
        const unsigned r   = gid - b * (unsigned)LN_;
        const unsigned l   = r >> 6;

        float p[E_];
        #pragma unroll
        for (int e = 0; e < E_; ++e) p[e] = stage[buf][e * TILE_ + tid];

        const float tv = __builtin_nontemporal_load(&target[b * (unsigned)LN_ + r]);

        // MAE term: sum |t - p_e| (unsorted; fw scaling applied at the end)
        float maes = 0.f;
        #pragma unroll
        for (int e = 0; e < E_; ++e) maes += fabsf(tv - p[e]);

        // Batcher odd-even mergesort network, 16 inputs, 63 comparators.
        #define CE(a,b) { float lo_ = fminf(p[a], p[b]); \
                          float hi_ = fmaxf(p[a], p[b]); \
                          p[a] = lo_; p[b] = hi_; }
        CE(0,1)  CE(2,3)  CE(4,5)   CE(6,7)   CE(8,9)   CE(10,11) CE(12,13) CE(14,15)
        CE(0,2)  CE(4,6)  CE(8,10)  CE(12,14) CE(1,3)   CE(5,7)   CE(9,11)  CE(13,15)
        CE(1,2)  CE(5,6)  CE(9,10)  CE(13,14)
        CE(0,4)  CE(8,12) CE(1,5)   CE(9,13)  CE(2,6)   CE(10,14) CE(3,7)   CE(11,15)
        CE(2,4)  CE(10,12) CE(3,5)  CE(11,13)
        CE(1,2)  CE(3,4)  CE(5,6)   CE(9,10)  CE(11,12) CE(13,14)
        CE(0,8)  CE(1,9)  CE(2,10)  CE(3,11)  CE(4,12)  CE(5,13)  CE(6,14)  CE(7,15)
        CE(4,8)  CE(5,9)  CE(6,10)  CE(7,11)
        CE(2,4)  CE(6,8)  CE(10,12) CE(3,5)   CE(7,9)   CE(11,13)
        CE(1,2)  CE(3,4)  CE(5,6)   CE(7,8)   CE(9,10)  CE(11,12) CE(13,14)
        #undef CE

        // dot = sum (2e-15) * sorted_p[e];  pair_sum = 2*dot; coef = -1/480
        float dot = 0.f;
        #pragma unroll
        for (int e = 0; e < E_; ++e) dot += (float)(2 * e - 15) * p[e];

        // kcrps = (mae_sum/16 - dot/240) * fw ; fw = featw[n]/64 >= 0
        const float kc = maes * (1.0f / 16.0f) - dot * (1.0f / 240.0f);
        acc += kc * fw * nodew[l];
    };

    // Software pipeline over this block's tiles (stride NBLK_ == gridDim.x).
    int buf = 0;
    issue((int)blockIdx.x, 0);
    for (int t = (int)blockIdx.x; t < NTILES_; t += NBLK_) {
        const int tn = t + NBLK_;
        if (tn < NTILES_) {
            issue(tn, buf ^ 1);                               // prefetch next tile
            asm volatile("s_wait_asynccnt 16" ::: "memory");  // tile t's 16 done
        } else {
            asm volatile("s_wait_asynccnt 0" ::: "memory");   // drain
        }
        compute(t, buf);
        buf ^= 1;
    }

    float bsum = block_reduce_256(acc);
    if (threadIdx.x == 0) partials[blockIdx.x] = bsum;
}

// -------- kernel 2: reduce partials, divide by (W * B) --------------
__global__ void __launch_bounds__(256)
k_finalize(const float* __restrict__ partials,
           const float* __restrict__ wsum,
           float* __restrict__ out) {
    float s = 0.f;
    for (int i = threadIdx.x; i < NBLK_; i += 256) s += partials[i];
    float t = block_reduce_256(s);
    if (threadIdx.x == 0) out[0] = t / (wsum[0] * (float)B_);
}

extern "C" void kernel_launch(void* const* d_in, const int* in_sizes, int n_in,
                              void* d_out, int out_size, void* d_ws, size_t ws_size,
                              hipStream_t stream) {
    (void)in_sizes; (void)n_in; (void)out_size; (void)ws_size;
    const float* pred   = (const float*)d_in[0];
    const float* target = (const float*)d_in[1];
    const float* nodew  = (const float*)d_in[2];
    const float* featw  = (const float*)d_in[3];
    float* ws       = (float*)d_ws;
    float* wsum     = ws;          // 1 float
    float* partials = ws + 256;    // NBLK_ floats (aligned region)
    float* out      = (float*)d_out;

    k_nodew_sum<<<1, 256, 0, stream>>>(nodew, wsum);
    k_crps_main<<<NBLK_, 256, 0, stream>>>(pred, target, nodew, featw, partials);
    k_finalize<<<1, 256, 0, stream>>>(partials, wsum, out);
}